// GAT_27960237097254
// MI455X (gfx1250) — compile-verified
//
#include <hip/hip_runtime.h>
#include <math.h>
#include <stdint.h>

// Problem constants (from reference)
static constexpr int  Bc  = 8;
static constexpr int  Nc  = 512;
static constexpr int  ENc = 256;
static constexpr int  EEc = 64;
static constexpr int  Hc  = 4;
static constexpr int  Lc  = 4;
static constexpr int  Dc  = 64;           // EN / H
static constexpr long Rr  = (long)Bc * Nc * Nc;   // 2,097,152 edge rows
static constexpr float NEGV  = -1.0e9f;
static constexpr float SLOPE = 0.2f;

typedef __attribute__((ext_vector_type(2))) float v2f;
typedef __attribute__((ext_vector_type(8))) float v8f;

// D = A(16x4 f32) * B(4x16 f32) + C(16x16 f32), wave32 CDNA5 WMMA
__device__ __forceinline__ v8f wmma4(v2f a, v2f b, v8f c) {
  return __builtin_amdgcn_wmma_f32_16x16x4_f32(false, a, false, b, (short)0, c,
                                               false, false);
}

__device__ __forceinline__ float wave_max(float v) {
  #pragma unroll
  for (int o = 16; o; o >>= 1) v = fmaxf(v, __shfl_xor(v, o, 32));
  return v;
}
__device__ __forceinline__ float wave_sum(float v) {
  #pragma unroll
  for (int o = 16; o; o >>= 1) v += __shfl_xor(v, o, 32);
  return v;
}

// CDNA5 async global->LDS copy (4B per lane), tracked by ASYNCcnt.
__device__ __forceinline__ void async_g2l_b32(const float* gsrc, float* lds_dst) {
  unsigned loff = (unsigned)(uintptr_t)lds_dst;   // LDS aperture: addr[31:0] = LDS offset
  asm volatile("global_load_async_to_lds_b32 %0, %1, off"
               :: "v"(loff), "v"(gsrc) : "memory");
}
__device__ __forceinline__ void wait_asynccnt0() {
  asm volatile("s_wait_asynccnt 0x0" ::: "memory");
}

// ---------------------------------------------------------------------------
// K0: ve[lh][f] = sum_d We[l,h,f,d]*a_edge[l,h,d];  veT[f][lh] transposed copy;
//     bedot[lh] = be[l,h,:].a_edge
// ---------------------------------------------------------------------------
__global__ void gat_prep(const float* __restrict__ We, const float* __restrict__ be,
                         const float* __restrict__ a_edge,
                         float* __restrict__ ve, float* __restrict__ veT,
                         float* __restrict__ bedot) {
  int t = blockIdx.x * blockDim.x + threadIdx.x;       // 0..1023
  if (t < 16 * 64) {
    int lh = t >> 6, f = t & 63;
    const float* w = We + ((long)lh * EEc + f) * Dc;
    const float* a = a_edge + (long)lh * Dc;
    float s = 0.f;
    #pragma unroll 8
    for (int d = 0; d < Dc; ++d) s += w[d] * a[d];
    ve[lh * 64 + f]  = s;
    veT[f * 16 + lh] = s;
    if (f == 0) {
      const float* bb = be + (long)lh * Dc;
      float s2 = 0.f;
      for (int d = 0; d < Dc; ++d) s2 += bb[d] * a[d];
      bedot[lh] = s2;
    }
  }
}

// ---------------------------------------------------------------------------
// Fused pass over edges (2M x 64 rows):
//   cols 0..63  -> out_edges = edges @ Wfe + bfe        (written if write_out)
//   cols 64..79 -> s_e[lh]   = edges . ve[lh] + bedot   (lh in [lhA, lhA+lhN))
// B matrix (64 x 80) staged in LDS via async global->LDS copies.
// One wave per 16-row tile, WMMA f32 16x16x4.
// ---------------------------------------------------------------------------
__global__ void gat_edges(const float* __restrict__ edges,
                          const float* __restrict__ Wfe, const float* __restrict__ bfe,
                          const float* __restrict__ veT, const float* __restrict__ bedot,
                          float* __restrict__ out_edges, float* __restrict__ se,
                          int lhA, int lhN, int write_out) {
  __shared__ float Bl[64][80];                         // 20 KB
  int t = threadIdx.x;                                 // 256 threads = 8 waves
  // async stage: Bl[k][0:64] = Wfe[k][:], Bl[k][64+lh] = veT[k*16+lh] = ve[lh][k]
  for (int i = t; i < 64 * 64; i += 256) {             // 16 copies/thread
    int k = i >> 6, c = i & 63;
    async_g2l_b32(Wfe + i, &Bl[k][c]);
  }
  for (int i = t; i < 64 * 16; i += 256) {             // 4 copies/thread
    int k = i >> 4, lh = i & 15;
    async_g2l_b32(veT + i, &Bl[k][64 + lh]);
  }
  wait_asynccnt0();
  __syncthreads();

  int wave = t >> 5, lane = t & 31;
  long rowTile = (long)blockIdx.x * 8 + wave;          // 131072 tiles total
  long r0 = rowTile * 16;
  int lrow = lane & 15;
  int khi  = (lane >> 4) * 2;                          // lanes 16-31 hold K+2,K+3
  const float* arow = edges + (r0 + lrow) * (long)EEc;

  v2f afr[16];                                         // whole 16x64 A tile in regs
  #pragma unroll
  for (int s = 0; s < 16; ++s) {
    int kk = 4 * s + khi;
    afr[s] = v2f{arow[kk], arow[kk + 1]};
  }

  for (int ct = 0; ct < 5; ++ct) {
    if (ct < 4 && !write_out) continue;                // skip out_edges work on replays
    int c = ct * 16 + lrow;
    float bias = (c < 64) ? bfe[c] : bedot[c - 64];
    v8f acc = {bias, bias, bias, bias, bias, bias, bias, bias};
    #pragma unroll
    for (int s = 0; s < 16; ++s) {
      int kk = 4 * s + khi;
      v2f bf = v2f{Bl[kk][c], Bl[kk + 1][c]};
      acc = wmma4(afr[s], bf, acc);
    }
    if (c < 64) {
      #pragma unroll
      for (int r = 0; r < 8; ++r) {
        long row = r0 + r + ((lane >= 16) ? 8 : 0);
        out_edges[row * EEc + c] = acc[r];
      }
    } else {
      int li = (c - 64) - lhA;
      if (li >= 0 && li < lhN) {
        #pragma unroll
        for (int r = 0; r < 8; ++r) {
          long row = r0 + r + ((lane >= 16) ? 8 : 0);
          se[(long)li * Rr + row] = acc[r];
        }
      }
    }
  }
}

// ---------------------------------------------------------------------------
// K1: h[b,head,n,d] = act(x)[b,n,:] @ Wn[l,head] + bn[l,head]   (act = ELU if l>0)
// One wave per 16x16 output tile. 4096 waves.
// ---------------------------------------------------------------------------
__global__ void gat_h(const float* __restrict__ x, const float* __restrict__ Wn,
                      const float* __restrict__ bn, float* __restrict__ h,
                      int l, int apply_elu) {
  int wid  = (blockIdx.x * blockDim.x + threadIdx.x) >> 5;   // 0..4095
  int lane = threadIdx.x & 31;
  int dt = wid & 3, nt = (wid >> 2) & 31, bh = wid >> 7;
  int b = bh >> 2, head = bh & 3;
  int lrow = lane & 15, khi = (lane >> 4) * 2;
  int n0 = nt * 16, dcol = dt * 16 + lrow;

  const float* xrow = x  + ((long)b * Nc + n0 + lrow) * ENc;
  const float* W    = Wn + (long)(l * Hc + head) * ENc * Dc;
  float bias = bn[(l * Hc + head) * Dc + dcol];
  v8f acc = {bias, bias, bias, bias, bias, bias, bias, bias};

  for (int k = 0; k < ENc; k += 4) {
    int kk = k + khi;
    float a0 = xrow[kk], a1 = xrow[kk + 1];
    if (apply_elu) {
      a0 = a0 > 0.f ? a0 : expm1f(a0);
      a1 = a1 > 0.f ? a1 : expm1f(a1);
    }
    v2f bf = v2f{W[(long)kk * Dc + dcol], W[(long)(kk + 1) * Dc + dcol]};
    acc = wmma4(v2f{a0, a1}, bf, acc);
  }
  float* hb = h + (long)bh * Nc * Dc;
  #pragma unroll
  for (int r = 0; r < 8; ++r) {
    int row = n0 + r + ((lane >= 16) ? 8 : 0);
    hb[(long)row * Dc + dcol] = acc[r];
  }
}

// ---------------------------------------------------------------------------
// K2: s_src/s_dst[bh,n] = h[bh,n,:].a_src / .a_dst  (thread per row, float4)
// ---------------------------------------------------------------------------
__global__ void gat_sdots(const float* __restrict__ h, const float* __restrict__ a_src,
                          const float* __restrict__ a_dst,
                          float* __restrict__ ssrc, float* __restrict__ sdst, int l) {
  int t = blockIdx.x * blockDim.x + threadIdx.x;       // B*H*N = 16384
  if (t >= Bc * Hc * Nc) return;
  int head = (t / Nc) & 3;
  const float4* hr = (const float4*)(h + (long)t * Dc);
  const float4* as = (const float4*)(a_src + (long)(l * Hc + head) * Dc);
  const float4* ad = (const float4*)(a_dst + (long)(l * Hc + head) * Dc);
  float s1 = 0.f, s2 = 0.f;
  #pragma unroll
  for (int i = 0; i < 16; ++i) {
    float4 v = hr[i], p = as[i], q = ad[i];
    s1 += v.x * p.x + v.y * p.y + v.z * p.z + v.w * p.w;
    s2 += v.x * q.x + v.y * q.y + v.z * q.z + v.w * q.w;
  }
  ssrc[t] = s1; sdst[t] = s2;
}

// ---------------------------------------------------------------------------
// K3: one block per (b, head, 16-row n tile): scores -> softmax -> alpha @ h.
// Scores (16x512 fp32, padded) live in LDS (~33 KB of the 320 KB WGP LDS).
// ---------------------------------------------------------------------------
__global__ void gat_attn(const float* __restrict__ seL, const float* __restrict__ ssrc,
                         const float* __restrict__ sdst,
                         const unsigned char* __restrict__ mask,   // jnp bool = 1 byte
                         const float* __restrict__ h, float* __restrict__ xout) {
  __shared__ float S[16][516];                         // pad: bank-conflict-free
  int blk = blockIdx.x;                                // 1024 blocks, 128 threads
  int nt = blk & 31, bh = blk >> 5, b = bh >> 2, head = bh & 3;
  int n0 = nt * 16;
  int t = threadIdx.x;

  const float* se = seL + (long)head * Rr + ((long)b * Nc + n0) * Nc;
  const float* ss = ssrc + (long)bh * Nc + n0;
  const float* sd = sdst + (long)bh * Nc;
  const unsigned char* mk = mask + (long)b * Nc;

  for (int i = t; i < 16 * Nc; i += 128) {
    int nr = i >> 9, m = i & 511;
    float sc = ss[nr] + sd[m] + se[(long)nr * Nc + m];
    sc = sc > 0.f ? sc : SLOPE * sc;                   // leaky_relu
    S[nr][m] = mk[m] ? sc : NEGV;
  }
  __syncthreads();

  int wave = t >> 5, lane = t & 31;
  for (int rr = 0; rr < 4; ++rr) {                     // 4 rows per wave
    int nr = wave * 4 + rr;
    float mx = -3.0e38f;
    for (int m = lane; m < Nc; m += 32) mx = fmaxf(mx, S[nr][m]);
    mx = wave_max(mx);
    float sum = 0.f;
    for (int m = lane; m < Nc; m += 32) { float e = expf(S[nr][m] - mx); S[nr][m] = e; sum += e; }
    sum = wave_sum(sum);
    float inv = 1.f / sum;
    for (int m = lane; m < Nc; m += 32) S[nr][m] *= inv;
  }
  __syncthreads();

  // out tile: wave w -> d columns [w*16, w*16+16); K = 512 over m
  int lrow = lane & 15, khi = (lane >> 4) * 2;
  int dcol = wave * 16 + lrow;
  const float* hb = h + (long)bh * Nc * Dc;
  v8f acc = {0.f, 0.f, 0.f, 0.f, 0.f, 0.f, 0.f, 0.f};
  for (int k = 0; k < Nc; k += 4) {
    int kk = k + khi;
    v2f af = v2f{S[lrow][kk], S[lrow][kk + 1]};
    v2f bf = v2f{hb[(long)kk * Dc + dcol], hb[(long)(kk + 1) * Dc + dcol]};
    acc = wmma4(af, bf, acc);
  }
  float* xo = xout + (long)b * Nc * ENc;
  #pragma unroll
  for (int r = 0; r < 8; ++r) {
    int row = n0 + r + ((lane >= 16) ? 8 : 0);
    xo[(long)row * ENc + head * Dc + dcol] = acc[r];
  }
}

// ---------------------------------------------------------------------------
// Final: out_nodes = x @ Wfn + bfn   ((4096 x 256) @ (256 x 256))
// ---------------------------------------------------------------------------
__global__ void gat_outnodes(const float* __restrict__ x, const float* __restrict__ Wfn,
                             const float* __restrict__ bfn, float* __restrict__ out) {
  int wid  = (blockIdx.x * blockDim.x + threadIdx.x) >> 5;   // 4096 waves
  int lane = threadIdx.x & 31;
  int ct = wid & 15, rt = wid >> 4;
  int lrow = lane & 15, khi = (lane >> 4) * 2;
  int r0 = rt * 16, c = ct * 16 + lrow;
  const float* xr = x + (long)(r0 + lrow) * ENc;
  float bias = bfn[c];
  v8f acc = {bias, bias, bias, bias, bias, bias, bias, bias};
  for (int k = 0; k < ENc; k += 4) {
    int kk = k + khi;
    v2f af = v2f{xr[kk], xr[kk + 1]};
    v2f bf = v2f{Wfn[(long)kk * ENc + c], Wfn[(long)(kk + 1) * ENc + c]};
    acc = wmma4(af, bf, acc);
  }
  #pragma unroll
  for (int r = 0; r < 8; ++r) {
    long row = r0 + r + ((lane >= 16) ? 8 : 0);
    out[row * ENc + c] = acc[r];
  }
}

// ---------------------------------------------------------------------------
extern "C" void kernel_launch(void* const* d_in, const int* in_sizes, int n_in,
                              void* d_out, int out_size, void* d_ws, size_t ws_size,
                              hipStream_t stream) {
  (void)in_sizes; (void)n_in; (void)out_size;
  const float* nodes  = (const float*)d_in[0];
  const float* edges  = (const float*)d_in[1];
  const unsigned char* mask = (const unsigned char*)d_in[2];
  const float* Wn     = (const float*)d_in[3];
  const float* bn     = (const float*)d_in[4];
  const float* We     = (const float*)d_in[5];
  const float* be     = (const float*)d_in[6];
  const float* a_src  = (const float*)d_in[7];
  const float* a_dst  = (const float*)d_in[8];
  const float* a_edge = (const float*)d_in[9];
  const float* Wfn    = (const float*)d_in[10];
  const float* bfn    = (const float*)d_in[11];
  const float* Wfe    = (const float*)d_in[12];
  const float* bfe    = (const float*)d_in[13];

  float* out_nodes = (float*)d_out;
  float* out_edges = out_nodes + (long)Bc * Nc * ENc;

  // workspace layout (float offsets)
  float* ws    = (float*)d_ws;
  float* ve    = ws;                       // 1024
  float* veT   = ws + 1024;                // 1024 (transposed ve for async staging)
  float* bedot = ws + 2048;                // 16
  float* ssrc  = ws + 2064;                // 16384
  float* sdst  = ws + 18448;               // 16384
  float* xA    = ws + 34832;               // 1,048,576
  float* xB    = xA + 1048576;
  float* hbuf  = xB + 1048576;
  float* sebuf = hbuf + 1048576;           // 16*R (all) or 4*R (per-layer)
  const size_t base = 3180560;
  bool allLayers = ws_size >= (base + (size_t)16 * Rr) * sizeof(float);

  gat_prep<<<4, 256, 0, stream>>>(We, be, a_edge, ve, veT, bedot);

  if (allLayers)  // single fused pass over edges: out_edges + s_e for all 16 (l,h)
    gat_edges<<<16384, 256, 0, stream>>>(edges, Wfe, bfe, veT, bedot,
                                         out_edges, sebuf, 0, 16, 1);

  const float* xin = nodes;
  float* xcur = xA;
  for (int l = 0; l < Lc; ++l) {
    if (!allLayers)
      gat_edges<<<16384, 256, 0, stream>>>(edges, Wfe, bfe, veT, bedot,
                                           out_edges, sebuf, l * Hc, Hc, l == 0 ? 1 : 0);
    gat_h<<<512, 256, 0, stream>>>(xin, Wn, bn, hbuf, l, l > 0 ? 1 : 0);
    gat_sdots<<<64, 256, 0, stream>>>(hbuf, a_src, a_dst, ssrc, sdst, l);
    const float* seL = allLayers ? (sebuf + (long)l * Hc * Rr) : sebuf;
    gat_attn<<<1024, 128, 0, stream>>>(seL, ssrc, sdst, mask, hbuf, xcur);
    xin = xcur;
    xcur = (xcur == xA) ? xB : xA;
  }
  gat_outnodes<<<512, 256, 0, stream>>>(xin, Wfn, bfn, out_nodes);
}